// PredictorModel_48790828482866
// MI455X (gfx1250) — compile-verified
//
#include <hip/hip_runtime.h>
#include <cstddef>

// ---------------------------------------------------------------------------
// Problem constants (from the reference)
// ---------------------------------------------------------------------------
#define NSAMP   2048
#define NRX     64
#define NTX     64
#define NENC    32
#define NBATCH  2
#define XPX     256
#define ZPX     256
#define NPIX    (XPX * ZPX)          // 65536
#define TIK     0.01f
#define R0_C    6.16e-3f             // DATA_SPAN0 / FS * C
#define DR_C    6.16e-5f             // C / FS
#define TWO_PI  6.283185307179586f

typedef float v2f __attribute__((ext_vector_type(2)));
typedef float v8f __attribute__((ext_vector_type(8)));

// ---------------------------------------------------------------------------
// Elementwise helpers
// ---------------------------------------------------------------------------
__global__ void k_real2complex(const float* __restrict__ src,
                               float2* __restrict__ dst, int n) {
    int i = blockIdx.x * 256 + threadIdx.x;
    if (i < n) dst[i] = make_float2(src[i], 0.0f);
}

__global__ void k_project_real(const float2* __restrict__ src,
                               float2* __restrict__ dst, int n) {
    int i = blockIdx.x * 256 + threadIdx.x;
    if (i < n) dst[i] = make_float2(src[i].x, 0.0f);
}

// hilbert filter multiply (in place), layout (b, f, chan) with chan = 1<<cshift
__global__ void k_hilbert(float2* __restrict__ x, int cshift, int n) {
    int i = blockIdx.x * 256 + threadIdx.x;
    if (i >= n) return;
    int f = (i >> cshift) & (NSAMP - 1);
    float h = (f == 0 || f == NSAMP / 2) ? 1.0f : (f < NSAMP / 2 ? 2.0f : 0.0f);
    float2 v = x[i];
    x[i] = make_float2(v.x * h, v.y * h);
}

// (b, s, r, t) -> (b, r, t, s)
__global__ void k_transpose(const float2* __restrict__ X,
                            float2* __restrict__ Y, int n) {
    int i = blockIdx.x * 256 + threadIdx.x;
    if (i >= n) return;
    int t = i & 63;
    int r = (i >> 6) & 63;
    int s = (i >> 12) & (NSAMP - 1);
    int b = i >> 23;
    Y[(((size_t)b * 64 + r) * 64 + t) * NSAMP + s] = X[i];
}

// ---------------------------------------------------------------------------
// Stockham radix-2 FFT stage; batch (channel) dim contiguous -> coalesced.
// sign = +1 forward, -1 inverse; scale folded into last stage for ifft.
// ---------------------------------------------------------------------------
__global__ void k_fft_stage(const float2* __restrict__ X, float2* __restrict__ Y,
                            int cshift, int p, float sign, float scale) {
    const int C = 1 << cshift;
    int tid = blockIdx.x * 256 + threadIdx.x;     // exactly C*1024 threads
    int c = tid & (C - 1);
    int q = tid >> cshift;                        // 0..1023 butterfly index
    size_t base = (size_t)blockIdx.y << (11 + cshift);
    int Lp = 1 << p;
    int j = q & (Lp - 1);
    int k = q >> p;
    float ang = sign * (-TWO_PI) * (float)j / (float)(Lp << 1);
    float sn, cs;
    __sincosf(ang, &sn, &cs);
    float2 a = X[base + ((size_t)(k * Lp + j) << cshift) + c];
    float2 b = X[base + ((size_t)(k * Lp + j + 1024) << cshift) + c];
    float2 wb = make_float2(cs * b.x - sn * b.y, cs * b.y + sn * b.x);
    Y[base + ((size_t)(2 * k * Lp + j) << cshift) + c] =
        make_float2((a.x + wb.x) * scale, (a.y + wb.y) * scale);
    Y[base + ((size_t)(2 * k * Lp + Lp + j) << cshift) + c] =
        make_float2((a.x - wb.x) * scale, (a.y - wb.y) * scale);
}

// ---------------------------------------------------------------------------
// H[f][t][e] = w[e][t] * exp(-2*pi*i * fftfreq(f) * delay[e][t])
// ---------------------------------------------------------------------------
__global__ void k_build_H(const float* __restrict__ delays,
                          const float* __restrict__ weights,
                          float2* __restrict__ H, int n) {
    int i = blockIdx.x * 256 + threadIdx.x;
    if (i >= n) return;
    int e = i & 31;
    int t = (i >> 5) & 63;
    int f = i >> 11;
    float freq = (float)((f < NSAMP / 2) ? f : f - NSAMP) / (float)NSAMP;
    float d = delays[e * 64 + t];
    float w = weights[e * 64 + t];
    float ang = -TWO_PI * freq * d;
    float sn, cs;
    __sincosf(ang, &sn, &cs);
    H[(size_t)f * (NTX * NENC) + t * NENC + e] = make_float2(w * cs, w * sn);
}

// ---------------------------------------------------------------------------
// Per-frequency Tikhonov inverse: Hinv(f) = (Hh H + lam I)^-1 Hh  (32x64)
// One block (128 threads) per frequency; Gauss-Jordan in LDS (SPD, no pivot).
// ---------------------------------------------------------------------------
__global__ void k_solve(const float2* __restrict__ H, float2* __restrict__ Hinv) {
    __shared__ float2 Hs[NTX * NENC];     // 16 KB
    __shared__ float2 A[32][96];          // 24 KB  [A | Hh]
    __shared__ float2 fac[32];
    __shared__ float2 piv;
    int f = blockIdx.x, tid = threadIdx.x;
    const float2* Hf = H + (size_t)f * (NTX * NENC);
    for (int i = tid; i < NTX * NENC; i += 128) Hs[i] = Hf[i];
    __syncthreads();
    // HtH + lam I
    for (int idx = tid; idx < 1024; idx += 128) {
        int e = idx >> 5, c2 = idx & 31;
        float sr = 0.f, si = 0.f;
        for (int t = 0; t < 64; ++t) {
            float2 he = Hs[t * 32 + e];
            float2 hc = Hs[t * 32 + c2];
            sr += he.x * hc.x + he.y * hc.y;   // conj(he)*hc
            si += he.x * hc.y - he.y * hc.x;
        }
        if (e == c2) sr += TIK;
        A[e][c2] = make_float2(sr, si);
    }
    // RHS: A[e][32+t] = conj(H[t][e])
    for (int idx = tid; idx < 2048; idx += 128) {
        int e = idx & 31, t = idx >> 5;
        float2 h = Hs[t * 32 + e];
        A[e][32 + t] = make_float2(h.x, -h.y);
    }
    __syncthreads();
    for (int k = 0; k < 32; ++k) {
        if (tid == 0) {
            float2 p = A[k][k];
            float d = p.x * p.x + p.y * p.y;
            piv = make_float2(p.x / d, -p.y / d);
        }
        __syncthreads();
        if (tid < 96) {
            float2 v = A[k][tid];
            A[k][tid] = make_float2(v.x * piv.x - v.y * piv.y,
                                    v.x * piv.y + v.y * piv.x);
        }
        __syncthreads();
        if (tid < 32) fac[tid] = A[tid][k];
        __syncthreads();
        for (int idx = tid; idx < 32 * 96; idx += 128) {
            int i2 = idx / 96, j2 = idx - i2 * 96;
            if (i2 != k) {
                float2 fkj = A[k][j2], fi = fac[i2];
                A[i2][j2].x -= fi.x * fkj.x - fi.y * fkj.y;
                A[i2][j2].y -= fi.x * fkj.y + fi.y * fkj.x;
            }
        }
        __syncthreads();
    }
    for (int idx = tid; idx < 2048; idx += 128) {
        int t = idx & 63, e = idx >> 6;
        Hinv[(size_t)f * (NENC * NTX) + e * 64 + t] = A[e][32 + t];
    }
}

// ---------------------------------------------------------------------------
// Encode: per (b,f)  E(64x32) = D(64x64) * H_f(64x32), complex fp32 via
// v_wmma_f32_16x16x4_f32.  8 waves/block, one 16x16 C-tile per wave.
// F32 WMMA has no A/B neg modifier (NEG must be 0): negate Aim in VALU.
// ---------------------------------------------------------------------------
__global__ void k_encode_wmma(const float2* __restrict__ D,
                              const float2* __restrict__ H,
                              float2* __restrict__ E) {
    const int lane = threadIdx.x & 31;
    const int wv = threadIdx.x >> 5;       // 0..7
    const int mt = wv & 3, nt = wv >> 2;   // 4 M-tiles x 2 N-tiles
    const int bf = blockIdx.x;             // b*2048 + f
    const float2* Dp = D + (size_t)bf * (NRX * NTX);
    const float2* Hp = H + (size_t)(bf & (NSAMP - 1)) * (NTX * NENC);
    const int row = lane & 15, kh = lane >> 4;
    const int r = mt * 16 + row;
    const int col = nt * 16 + row;
    v8f cre = {}; v8f cim = {};
    for (int k = 0; k < 16; ++k) {
        int t0 = k * 4 + kh * 2;
        float2 a0 = Dp[r * 64 + t0], a1 = Dp[r * 64 + t0 + 1];
        float2 b0 = Hp[t0 * 32 + col], b1 = Hp[(t0 + 1) * 32 + col];
        v2f are  = {a0.x, a1.x};
        v2f aim  = {a0.y, a1.y};
        v2f naim = {-a0.y, -a1.y};
        v2f bre  = {b0.x, b1.x};
        v2f bim  = {b0.y, b1.y};
        cre = __builtin_amdgcn_wmma_f32_16x16x4_f32(false, are,  false, bre, (short)0, cre, false, false);
        cre = __builtin_amdgcn_wmma_f32_16x16x4_f32(false, naim, false, bim, (short)0, cre, false, false);
        cim = __builtin_amdgcn_wmma_f32_16x16x4_f32(false, are,  false, bim, (short)0, cim, false, false);
        cim = __builtin_amdgcn_wmma_f32_16x16x4_f32(false, aim,  false, bre, (short)0, cim, false, false);
    }
    float2* Ep = E + (size_t)bf * (NRX * NENC);
    for (int i = 0; i < 8; ++i) {
        int rr = mt * 16 + i + 8 * kh;
        Ep[rr * 32 + col] = make_float2(cre[i], cim[i]);
    }
}

// ---------------------------------------------------------------------------
// Decode: per (b,f)  G(64x64) = De(64x32) * Hinv_f(32x64).  16 waves/block.
// ---------------------------------------------------------------------------
__global__ void k_decode_wmma(const float2* __restrict__ E,
                              const float2* __restrict__ Hi,
                              float2* __restrict__ G) {
    const int lane = threadIdx.x & 31;
    const int wv = threadIdx.x >> 5;       // 0..15
    const int mt = wv & 3, nt = wv >> 2;   // 4 x 4 tiles
    const int bf = blockIdx.x;
    const float2* Ep = E + (size_t)bf * (NRX * NENC);
    const float2* Hp = Hi + (size_t)(bf & (NSAMP - 1)) * (NENC * NTX);
    const int row = lane & 15, kh = lane >> 4;
    const int r = mt * 16 + row;
    const int col = nt * 16 + row;
    v8f cre = {}; v8f cim = {};
    for (int k = 0; k < 8; ++k) {
        int e0 = k * 4 + kh * 2;
        float2 a0 = Ep[r * 32 + e0], a1 = Ep[r * 32 + e0 + 1];
        float2 b0 = Hp[e0 * 64 + col], b1 = Hp[(e0 + 1) * 64 + col];
        v2f are  = {a0.x, a1.x};
        v2f aim  = {a0.y, a1.y};
        v2f naim = {-a0.y, -a1.y};
        v2f bre  = {b0.x, b1.x};
        v2f bim  = {b0.y, b1.y};
        cre = __builtin_amdgcn_wmma_f32_16x16x4_f32(false, are,  false, bre, (short)0, cre, false, false);
        cre = __builtin_amdgcn_wmma_f32_16x16x4_f32(false, naim, false, bim, (short)0, cre, false, false);
        cim = __builtin_amdgcn_wmma_f32_16x16x4_f32(false, are,  false, bim, (short)0, cim, false, false);
        cim = __builtin_amdgcn_wmma_f32_16x16x4_f32(false, aim,  false, bre, (short)0, cim, false, false);
    }
    float2* Gp = G + (size_t)bf * (NRX * NTX);
    for (int i = 0; i < 8; ++i) {
        int rr = mt * 16 + i + 8 * kh;
        Gp[rr * 64 + col] = make_float2(cre[i], cim[i]);
    }
}

// ---------------------------------------------------------------------------
// bf[e][pix] = distance(element e, pixel)
// ---------------------------------------------------------------------------
__global__ void k_bf_delays(const float* __restrict__ rxpos,
                            float* __restrict__ bf, int n) {
    int i = blockIdx.x * 256 + threadIdx.x;
    if (i >= n) return;
    int pix = i & (NPIX - 1);
    int e = i >> 16;
    int xi = pix & 255, zi = pix >> 8;
    float x = (-15.0f + (float)xi * (30.0f / 255.0f)) * 1e-3f;
    float z = (10.0f + (float)zi * (40.0f / 255.0f)) * 1e-3f;
    float dx = x - rxpos[e];
    bf[i] = sqrtf(dx * dx + z * z);
}

// ---------------------------------------------------------------------------
// Delay-and-sum beamform.  iq layout (b, r, t, s); one thread per pixel,
// full tx/rx loops -> no atomics, writes |img|+eps directly.
// ---------------------------------------------------------------------------
__global__ void k_beamform(const float2* __restrict__ iq,
                           const float* __restrict__ bf,
                           float* __restrict__ mag) {
    int pix = blockIdx.x * 256 + threadIdx.x;
    int b = blockIdx.y;
    const float invDR = 1.0f / DR_C;
    float accr = 0.f, acci = 0.f;
    for (int t = 0; t < NTX; ++t) {
        float dt = bf[t * NPIX + pix];
        size_t base_t = ((size_t)b * 64 * 64 + t) * NSAMP;
        for (int r = 0; r < NRX; ++r) {
            float dr = bf[r * NPIX + pix];
            float pos = (dt + dr - R0_C) * invDR;
            float fi = floorf(pos);
            int i0 = (int)fi;
            float frac = pos - fi;
            if (i0 >= 0 && i0 < NSAMP - 1) {
                const float2* sp = iq + base_t + (size_t)r * 64 * NSAMP + i0;
                float2 v0 = sp[0], v1 = sp[1];
                accr += v0.x + (v1.x - v0.x) * frac;
                acci += v0.y + (v1.y - v0.y) * frac;
            }
        }
    }
    mag[b * NPIX + pix] = sqrtf(accr * accr + acci * acci) + 1e-15f;
}

__global__ void k_max_reduce(const float* __restrict__ mag, float* __restrict__ mx) {
    __shared__ float s[256];
    int b = blockIdx.x;
    float m = 0.f;
    for (int i = threadIdx.x; i < NPIX; i += 256) m = fmaxf(m, mag[b * NPIX + i]);
    s[threadIdx.x] = m;
    __syncthreads();
    for (int o = 128; o > 0; o >>= 1) {
        if (threadIdx.x < o) s[threadIdx.x] = fmaxf(s[threadIdx.x], s[threadIdx.x + o]);
        __syncthreads();
    }
    if (threadIdx.x == 0) mx[b] = s[0];
}

__global__ void k_finalize(const float* __restrict__ mag,
                           const float* __restrict__ mx,
                           float* __restrict__ out, int n) {
    int i = blockIdx.x * 256 + threadIdx.x;
    if (i >= n) return;
    int b = i >> 16;
    float v = 20.0f * log10f(mag[i] / mx[b]);
    out[i] = fminf(0.0f, fmaxf(-60.0f, v));
}

// ---------------------------------------------------------------------------
// Host side
// ---------------------------------------------------------------------------
extern "C" void kernel_launch(void* const* d_in, const int* in_sizes, int n_in,
                              void* d_out, int out_size, void* d_ws, size_t ws_size,
                              hipStream_t stream) {
    (void)in_sizes; (void)n_in; (void)out_size; (void)ws_size;
    const float* datas   = (const float*)d_in[0];   // (2,2048,64,64)
    const float* delays  = (const float*)d_in[2];   // (32,64)
    const float* weights = (const float*)d_in[3];   // (32,64)
    const float* rxpos   = (const float*)d_in[4];   // (64,)
    float* out = (float*)d_out;

    // ---- workspace carving -------------------------------------------------
    const size_t SZBIG = (size_t)NBATCH * NSAMP * NRX * NTX;   // complex elems
    char* ws = (char*)d_ws;
    float2* bufA = (float2*)ws;                       ws += SZBIG * sizeof(float2);
    float2* bufB = (float2*)ws;                       ws += SZBIG * sizeof(float2);
    float2* H    = (float2*)ws;                       ws += (size_t)NSAMP * NTX * NENC * sizeof(float2);
    float2* Hinv = (float2*)ws;                       ws += (size_t)NSAMP * NENC * NTX * sizeof(float2);
    float*  bfD  = (float*)ws;                        ws += (size_t)64 * NPIX * sizeof(float);
    float*  mag  = (float*)ws;                        ws += (size_t)NBATCH * NPIX * sizeof(float);
    float*  mxv  = (float*)ws;

    const int N64 = NBATCH * NSAMP * NRX * NTX;    // 16777216 (width-64 elems)
    const int N32 = NBATCH * NSAMP * NRX * NENC;   // 8388608  (width-32 elems)
    const int NBF = NSAMP * NTX * NENC;            // 4194304

    float2 *cur = bufA, *alt = bufB;
    auto swapbuf = [&]() { float2* t = cur; cur = alt; alt = t; };

    auto fft_pass = [&](int cshift, float sign, bool invscale) {
        dim3 grid(((1u << cshift) * 1024) / 256, NBATCH, 1);
        for (int p = 0; p < 11; ++p) {
            float scale = (invscale && p == 10) ? (1.0f / (float)NSAMP) : 1.0f;
            k_fft_stage<<<grid, 256, 0, stream>>>(cur, alt, cshift, p, sign, scale);
            swapbuf();
        }
    };

    // ---- setup: H, Hinv, bf delays ----------------------------------------
    k_build_H<<<NBF / 256, 256, 0, stream>>>(delays, weights, H, NBF);
    k_solve<<<NSAMP, 128, 0, stream>>>(H, Hinv);
    k_bf_delays<<<(64 * NPIX) / 256, 256, 0, stream>>>(rxpos, bfD, 64 * NPIX);

    // ---- D = fft(datas) ----------------------------------------------------
    k_real2complex<<<N64 / 256, 256, 0, stream>>>(datas, cur, N64);
    fft_pass(12, 1.0f, false);

    // ---- encode (WMMA), ifft, re, fft -------------------------------------
    k_encode_wmma<<<NBATCH * NSAMP, 256, 0, stream>>>(cur, H, alt);
    swapbuf();
    fft_pass(11, -1.0f, true);
    k_project_real<<<N32 / 256, 256, 0, stream>>>(cur, alt, N32);
    swapbuf();
    fft_pass(11, 1.0f, false);

    // ---- decode (WMMA), ifft, re ------------------------------------------
    k_decode_wmma<<<NBATCH * NSAMP, 512, 0, stream>>>(cur, Hinv, alt);
    swapbuf();
    fft_pass(12, -1.0f, true);
    k_project_real<<<N64 / 256, 256, 0, stream>>>(cur, alt, N64);
    swapbuf();

    // ---- hilbert: fft, filter, ifft ---------------------------------------
    fft_pass(12, 1.0f, false);
    k_hilbert<<<N64 / 256, 256, 0, stream>>>(cur, 12, N64);
    fft_pass(12, -1.0f, true);

    // ---- transpose to (b,r,t,s) and beamform ------------------------------
    k_transpose<<<N64 / 256, 256, 0, stream>>>(cur, alt, N64);
    dim3 bgrid(NPIX / 256, NBATCH, 1);
    k_beamform<<<bgrid, 256, 0, stream>>>(alt, bfD, mag);

    // ---- log compression ---------------------------------------------------
    k_max_reduce<<<NBATCH, 256, 0, stream>>>(mag, mxv);
    k_finalize<<<(NBATCH * NPIX) / 256, 256, 0, stream>>>(mag, mxv, out, NBATCH * NPIX);
}